// SegmentationShader_6528350290142
// MI455X (gfx1250) — compile-verified
//
#include <hip/hip_runtime.h>
#include <stdint.h>

// Problem constants (from reference):
#define BM   8
#define VV   6890
#define CC   16
#define FF   13776
#define FTOT (BM * FF)            // 110208 faces total
#define NPIX (BM * 512 * 512)     // 2097152 pixels (K==1)

typedef int   v4i __attribute__((ext_vector_type(4)));

// ---------------------------------------------------------------------------
// Kernel A: collapse the double indirection.
//   tab[f] = { verts_class[faces[f][0], 0], verts_class[faces[f][1], 0],
//              verts_class[faces[f][2], 0], 0 }
// 1.76 MB table -> L2-resident on MI455X (192 MB L2).
// ---------------------------------------------------------------------------
__global__ __launch_bounds__(256) void build_face_tab(
    const float* __restrict__ vc,       // [B*V, C] packed, channel stride CC
    const int*   __restrict__ faces,    // [FTOT, 3]
    float4*      __restrict__ tab)      // [FTOT]
{
    int f = blockIdx.x * 256 + threadIdx.x;
    if (f >= FTOT) return;
    int i0 = faces[3 * f + 0];
    int i1 = faces[3 * f + 1];
    int i2 = faces[3 * f + 2];
    float4 t;
    t.x = vc[i0 * CC];   // channel 0 only — output keeps only c==0
    t.y = vc[i1 * CC];
    t.z = vc[i2 * CC];
    t.w = 0.0f;
    tab[f] = t;
}

// ---------------------------------------------------------------------------
// Kernel B: per-pixel interpolation. 4 pixels / thread.
// Streaming operands staged LDS-side via gfx1250 async load-to-LDS
// (ASYNCcnt path), gather served from the compact L2-resident face table.
// ---------------------------------------------------------------------------
__global__ __launch_bounds__(256) void seg_shader(
    const int*    __restrict__ p2f,     // [NPIX]
    const float*  __restrict__ bary,    // [NPIX, 3]
    const float4* __restrict__ tab,     // [FTOT]
    int*          __restrict__ out)     // [NPIX] int32
{
    __shared__ __align__(16) int   s_p2f[256 * 4];     //  4 KB
    __shared__ __align__(16) float s_bary[256 * 12];   // 12 KB

    const int tid = threadIdx.x;
    const int gid = blockIdx.x * 256 + tid;            // thread = 4 pixels

    // LDS byte offsets (flat->LDS: low 32 bits of the shared aperture addr)
    uint32_t lds_p = (uint32_t)(uintptr_t)(&s_p2f[tid * 4]);
    uint32_t lds_b = (uint32_t)(uintptr_t)(&s_bary[tid * 12]);
    // Per-lane global byte offsets (GVS mode: saddr64 + vaddr32 + inst_offset)
    uint32_t vof_p = (uint32_t)gid * 16u;   // 4 ints
    uint32_t vof_b = (uint32_t)gid * 48u;   // 12 floats
    uint64_t sad_p = (uint64_t)(uintptr_t)p2f;
    uint64_t sad_b = (uint64_t)(uintptr_t)bary;

    // ISA 15.18: inst_offset is added to BOTH the LDS and the global address,
    // so offset:0/16/32 walks the 48B bary tile with a single LDS base VGPR.
    asm volatile(
        "global_load_async_to_lds_b128 %0, %2, %4 offset:0\n\t"
        "global_load_async_to_lds_b128 %1, %3, %5 offset:0\n\t"
        "global_load_async_to_lds_b128 %1, %3, %5 offset:16\n\t"
        "global_load_async_to_lds_b128 %1, %3, %5 offset:32\n\t"
        "s_wait_asynccnt 0x0"
        :
        : "v"(lds_p), "v"(lds_b), "v"(vof_p), "v"(vof_b),
          "s"(sad_p), "s"(sad_b)
        : "memory");

    int res[4];
#pragma unroll
    for (int j = 0; j < 4; ++j) {
        int p = s_p2f[tid * 4 + j];
        int v = 0;
        if (p >= 0) {
            float4 fc = tab[p];                    // single b128 L2 gather
            float b0 = s_bary[tid * 12 + 3 * j + 0];
            float b1 = s_bary[tid * 12 + 3 * j + 1];
            float b2 = s_bary[tid * 12 + 3 * j + 2];
            v = (int)(b0 * fc.x + b1 * fc.y + b2 * fc.z);  // trunc == astype(i32)
        }
        res[j] = v;
    }
    v4i r; r.x = res[0]; r.y = res[1]; r.z = res[2]; r.w = res[3];
    __builtin_nontemporal_store(r, (v4i*)(out + gid * 4));  // no reuse
}

// ---------------------------------------------------------------------------
// Inputs (setup_inputs order): verts_class f32 [B,V,C], faces i32 [FTOT,3],
// pix_to_face i32 [B,H,W,K], bary_coords f32 [B,H,W,K,3].
// Output: int32 [B,H,W,K] = NPIX elements.
// ---------------------------------------------------------------------------
extern "C" void kernel_launch(void* const* d_in, const int* in_sizes, int n_in,
                              void* d_out, int out_size, void* d_ws, size_t ws_size,
                              hipStream_t stream)
{
    const float* vc    = (const float*)d_in[0];
    const int*   faces = (const int*)  d_in[1];
    const int*   p2f   = (const int*)  d_in[2];
    const float* bary  = (const float*)d_in[3];

    float4* tab = (float4*)d_ws;          // FTOT * 16 B = 1.76 MB scratch
    int*    out = (int*)d_out;

    build_face_tab<<<(FTOT + 255) / 256, 256, 0, stream>>>(vc, faces, tab);
    seg_shader<<<NPIX / 4 / 256, 256, 0, stream>>>(p2f, bary, tab, out);
}